// STGAT_61581241090090
// MI455X (gfx1250) — compile-verified
//
#include <hip/hip_runtime.h>
#include <stdint.h>

// ---------------------------------------------------------------------------
// ST-GAT: gat1(H=4,Cph=32) -> ELU -> gat2(H=1,C=128) -> TCN(k=3,len=4) -> LN
// fp32 end-to-end. All dense work (both GAT GEMMs and the im2col'd TCN) runs
// on V_WMMA_F32_16X16X4_F32. Edge softmax/aggregate phase is the true
// bottleneck (L2-resident gathers + global_atomic_add_f32) and is mapped as
// wave-per-edge with float4 gathers.
// ---------------------------------------------------------------------------

typedef __attribute__((ext_vector_type(2)))  float v2f;
typedef __attribute__((ext_vector_type(8)))  float v8f;

#define LEAKY 0.2f
#define SM_EPS 1e-16f
#define LN_EPS 1e-5f

__device__ __forceinline__ v8f wmma4(v2f a, v2f b, v8f c) {
    return __builtin_amdgcn_wmma_f32_16x16x4_f32(
        false, a, false, b, (short)0, c, false, false);
}

__device__ __forceinline__ v2f ldv2g(const float* p) {     // global float2
    const float2 t = *(const float2*)p;
    v2f r; r.x = t.x; r.y = t.y; return r;
}
__device__ __forceinline__ v2f ldv2s(const float* p) {     // LDS float2
    const float2 t = *(const float2*)p;
    v2f r; r.x = t.x; r.y = t.y; return r;
}

// ---------------- utility ----------------

__global__ void fillf_kernel(float* __restrict__ p, int n, float v) {
    const int i = blockIdx.x * blockDim.x + threadIdx.x;
    if (i < n) p[i] = v;
}

// W[K,C] -> Wt[C,K] so GEMM B-fragments are contiguous float2 per lane.
__global__ void transpose_kernel(const float* __restrict__ W, float* __restrict__ Wt,
                                 int K, int C) {
    const int idx = blockIdx.x * blockDim.x + threadIdx.x;
    if (idx >= K * C) return;
    const int k = idx / C, c = idx - k * C;
    Wt[c * K + k] = W[idx];
}

__device__ __forceinline__ float atomicMaxFloat(float* addr, float value) {
    // IEEE ordering trick (init must be a plain negative float, we use -FLT_MAX)
    if (value >= 0.0f)
        return __int_as_float(atomicMax((int*)addr, __float_as_int(value)));
    else
        return __uint_as_float(atomicMin((unsigned int*)addr, __float_as_uint(value)));
}

// ---------------- WMMA GEMM: D[M,128] = A[M,K] @ Wt[128,K]^T ----------------
// One wave computes a 16x64 strip: A fragment reused across 4 N-tiles.
// K is compile-time (64 or 128); all indexing is int32.

template<int K>
__global__ __launch_bounds__(32)
void gemm_wmma_kernel(const float* __restrict__ A, const float* __restrict__ Bt,
                      float* __restrict__ D) {
    const int colBlk = blockIdx.x * 64;            // 2 blocks cover 128 cols
    const int row0   = blockIdx.y * 16;
    const int lane = threadIdx.x, half = lane >> 4, l15 = lane & 15;

    v8f acc0 = {}, acc1 = {}, acc2 = {}, acc3 = {};
    const float* arow = A  + (row0 + l15) * K;
    const float* bp0  = Bt + (colBlk + l15) * K;       // contiguous per lane
    const float* bp1  = bp0 + 16 * K;
    const float* bp2  = bp0 + 32 * K;
    const float* bp3  = bp0 + 48 * K;

    #pragma unroll 4
    for (int k = 0; k < K; k += 4) {
        const int ko = k + half * 2;                   // lanes16-31 take K+{2,3}
        const v2f a = ldv2g(arow + ko);
        acc0 = wmma4(a, ldv2g(bp0 + ko), acc0);
        acc1 = wmma4(a, ldv2g(bp1 + ko), acc1);
        acc2 = wmma4(a, ldv2g(bp2 + ko), acc2);
        acc3 = wmma4(a, ldv2g(bp3 + ko), acc3);
    }

    // C/D layout: VGPR r -> row r + half*8, col = lane&15
    float* drow = D + (row0 + half * 8) * 128 + colBlk + l15;
    #pragma unroll
    for (int r = 0; r < 8; ++r) {
        drow[r * 128 +  0] = acc0[r];
        drow[r * 128 + 16] = acc1[r];
        drow[r * 128 + 32] = acc2[r];
        drow[r * 128 + 48] = acc3[r];
    }
}

// ---------------- attention dots ----------------

__global__ __launch_bounds__(32)
void attn_dots_kernel(const float* __restrict__ xp,
                      const float* __restrict__ att_s, const float* __restrict__ att_d,
                      float* __restrict__ as_, float* __restrict__ ad_,
                      int H, int Cph) {
    const int n    = blockIdx.x;
    const int lane = threadIdx.x;
    const float* row = xp + n * 128;
    float vs[4], vd[4];
    #pragma unroll
    for (int j = 0; j < 4; ++j) {
        const int c = j * 32 + lane;
        const float xv = row[c];
        vs[j] = xv * att_s[c];
        vd[j] = xv * att_d[c];
    }
    if (Cph == 32) {                       // H=4: channel block j == head j
        #pragma unroll
        for (int j = 0; j < 4; ++j) {
            float s = vs[j], d = vd[j];
            for (int off = 16; off; off >>= 1) {
                s += __shfl_xor(s, off, 32);
                d += __shfl_xor(d, off, 32);
            }
            if (lane == 0) { as_[n * 4 + j] = s; ad_[n * 4 + j] = d; }
        }
    } else {                               // H=1, Cph=128
        float s = vs[0] + vs[1] + vs[2] + vs[3];
        float d = vd[0] + vd[1] + vd[2] + vd[3];
        for (int off = 16; off; off >>= 1) {
            s += __shfl_xor(s, off, 32);
            d += __shfl_xor(d, off, 32);
        }
        if (lane == 0) { as_[n] = s; ad_[n] = d; }
    }
}

// ---------------- edge passes (softmax grouped by dst) ----------------
// Templated on H so e = t/H is a shift; all int32 indexing.

__device__ __forceinline__ void edge_nodes(int e, int E,
                                           const long long* __restrict__ esrc,
                                           const long long* __restrict__ edst,
                                           int& s, int& d) {
    if (e < E) { s = (int)esrc[e]; d = (int)edst[e]; }
    else       { s = d = e - E; }          // appended self-loops
}

template<int H>
__global__ __launch_bounds__(256)
void edge_alpha_max_kernel(const long long* __restrict__ esrc, const long long* __restrict__ edst,
                           const float* __restrict__ as_, const float* __restrict__ ad_,
                           float* __restrict__ amax, int E, int Etot) {
    const int t = blockIdx.x * 256 + threadIdx.x;
    if (t >= Etot * H) return;
    const int e = t / H, h = t - e * H;
    int s, d; edge_nodes(e, E, esrc, edst, s, d);
    float a = as_[s * H + h] + ad_[d * H + h];
    a = a > 0.0f ? a : LEAKY * a;
    atomicMaxFloat(&amax[d * H + h], a);
}

template<int H>
__global__ __launch_bounds__(256)
void edge_expsum_kernel(const long long* __restrict__ esrc, const long long* __restrict__ edst,
                        const float* __restrict__ as_, const float* __restrict__ ad_,
                        const float* __restrict__ amax, float* __restrict__ denom,
                        int E, int Etot) {
    const int t = blockIdx.x * 256 + threadIdx.x;
    if (t >= Etot * H) return;
    const int e = t / H, h = t - e * H;
    int s, d; edge_nodes(e, E, esrc, edst, s, d);
    float a = as_[s * H + h] + ad_[d * H + h];
    a = a > 0.0f ? a : LEAKY * a;
    atomicAdd(&denom[d * H + h], expf(a - amax[d * H + h]));
}

// One wave per edge; each lane moves 4 of 128 channels (float4 gather,
// 4x global_atomic_add_f32 scatter). Prefetch covers the gather behind the
// softmax-coefficient scalar math.
template<int H>
__global__ __launch_bounds__(256)
void edge_agg_kernel(const long long* __restrict__ esrc, const long long* __restrict__ edst,
                     const float* __restrict__ as_, const float* __restrict__ ad_,
                     const float* __restrict__ amax, const float* __restrict__ denom,
                     const float* __restrict__ xp, float* __restrict__ out,
                     int E, int Etot) {
    const int t = blockIdx.x * 256 + threadIdx.x;
    const int e = t >> 5;
    if (e >= Etot) return;
    const int lane = threadIdx.x & 31;
    int s, d; edge_nodes(e, E, esrc, edst, s, d);
    const int c0 = lane * 4;                  // 4 consecutive channels, same head
    constexpr int Cph = 128 / H;
    const int h = c0 / Cph;
    __builtin_prefetch(xp + s * 128 + c0, 0, 0);   // global_prefetch_b8
    float a = as_[s * H + h] + ad_[d * H + h];
    a = a > 0.0f ? a : LEAKY * a;
    const float coef = expf(a - amax[d * H + h]) / (denom[d * H + h] + SM_EPS);
    const float4 xv = *(const float4*)(xp + s * 128 + c0);
    float* o = out + d * 128 + c0;
    atomicAdd(o + 0, xv.x * coef);
    atomicAdd(o + 1, xv.y * coef);
    atomicAdd(o + 2, xv.z * coef);
    atomicAdd(o + 3, xv.w * coef);
}

// ---------------- bias + ELU ----------------

__global__ __launch_bounds__(256)
void bias_elu_kernel(float* __restrict__ h, const float* __restrict__ b, int n) {
    const int i = blockIdx.x * 256 + threadIdx.x;
    if (i >= n) return;
    float v = h[i] + b[i & 127];
    h[i] = v > 0.0f ? v : expf(v) - 1.0f;
}

// ---------------- fused TCN (im2col WMMA GEMM) + LayerNorm ------------------
// Block = 128 threads (4 waves) handles 16 nodes (= 4 length-4 time blocks).
// im2col A[16 x 384] built in LDS (K ordered kk = i*3 + kq so the transposed
// weight matrix B[kk][c] = tw[c*384 + kk] is a contiguous per-lane stream).
// Each wave computes two 16x16 output tiles via fp32 WMMA, then does the
// per-node LayerNorm with wave32 shuffle reductions.

__global__ __launch_bounds__(128)
void tcn_ln_kernel(const float* __restrict__ h2, const float* __restrict__ b2,
                   const float* __restrict__ tw,   // [O=128][I=128][K=3]
                   const float* __restrict__ tb,
                   const float* __restrict__ g, const float* __restrict__ be,
                   float* __restrict__ out) {
    __shared__ float s_in[16][128];
    __shared__ float sA[16][388];     // 388 % 64 = 4 -> conflict-free tr reads
    __shared__ float sy[16][132];     // padded for tile stores

    const int base = blockIdx.x * 16;
    const int tid  = threadIdx.x;
    const float b2v = b2[tid];

    #pragma unroll
    for (int m = 0; m < 16; ++m)
        s_in[m][tid] = h2[(base + m) * 128 + tid] + b2v;
    __syncthreads();

    // im2col: sA[m][i*3+kq] = in[blk, t+kq-1][i], zero outside the 4-block
    #pragma unroll
    for (int m = 0; m < 16; ++m) {
        const int tloc = m & 3, blk = m & ~3;
        #pragma unroll
        for (int kq = 0; kq < 3; ++kq) {
            const int tt = tloc + kq - 1;
            sA[m][tid * 3 + kq] = (tt >= 0 && tt < 4) ? s_in[blk + tt][tid] : 0.0f;
        }
    }
    __syncthreads();

    const int wv = tid >> 5, lane = tid & 31;
    const int half = lane >> 4, l15 = lane & 15;
    const int cb0 = wv * 32;                     // wave covers 2 N-tiles

    v8f acc0 = {}, acc1 = {};
    const float* bq0 = tw + (cb0 + l15) * 384;   // contiguous weight stream
    const float* bq1 = bq0 + 16 * 384;
    const float* ar  = &sA[l15][0];

    #pragma unroll 8
    for (int k = 0; k < 384; k += 4) {
        const int ko = k + half * 2;
        const v2f a = ldv2s(ar + ko);            // ds_load_b64
        acc0 = wmma4(a, ldv2g(bq0 + ko), acc0);
        acc1 = wmma4(a, ldv2g(bq1 + ko), acc1);
    }

    #pragma unroll
    for (int r = 0; r < 8; ++r) {
        sy[r + half * 8][cb0 + l15]      = acc0[r];
        sy[r + half * 8][cb0 + 16 + l15] = acc1[r];
    }
    __syncthreads();

    // LayerNorm: wave wv handles nodes wv*4 .. wv*4+3
    float tbv[4], gv[4], bev[4];
    #pragma unroll
    for (int j = 0; j < 4; ++j) {
        const int cc = lane + 32 * j;
        tbv[j] = tb[cc]; gv[j] = g[cc]; bev[j] = be[cc];
    }
    for (int q = 0; q < 4; ++q) {
        const int nn = wv * 4 + q;
        float yv[4], s = 0.0f, ss = 0.0f;
        #pragma unroll
        for (int j = 0; j < 4; ++j) {
            yv[j] = sy[nn][lane + 32 * j] + tbv[j];
            s += yv[j]; ss += yv[j] * yv[j];
        }
        for (int off = 16; off; off >>= 1) {
            s  += __shfl_xor(s,  off, 32);
            ss += __shfl_xor(ss, off, 32);
        }
        const float mu  = s * (1.0f / 128.0f);
        const float var = ss * (1.0f / 128.0f) - mu * mu;
        const float rs  = rsqrtf(var + LN_EPS);
        #pragma unroll
        for (int j = 0; j < 4; ++j) {
            const int cc = lane + 32 * j;
            out[(base + nn) * 128 + cc] = (yv[j] - mu) * rs * gv[j] + bev[j];
        }
    }
}

// ---------------- host-side orchestration ----------------

static inline unsigned gblk(long long n, int b) { return (unsigned)((n + b - 1) / b); }

extern "C" void kernel_launch(void* const* d_in, const int* in_sizes, int n_in,
                              void* d_out, int out_size, void* d_ws, size_t ws_size,
                              hipStream_t stream) {
    const float*     x     = (const float*)d_in[0];
    const long long* ei    = (const long long*)d_in[1];   // int64 [2, E]
    const float*     W1    = (const float*)d_in[2];
    const float*     atts1 = (const float*)d_in[3];
    const float*     attd1 = (const float*)d_in[4];
    const float*     b1    = (const float*)d_in[5];
    const float*     W2    = (const float*)d_in[6];
    const float*     atts2 = (const float*)d_in[7];
    const float*     attd2 = (const float*)d_in[8];
    const float*     b2    = (const float*)d_in[9];
    const float*     tw    = (const float*)d_in[10];
    const float*     tb    = (const float*)d_in[11];
    const float*     lg    = (const float*)d_in[12];
    const float*     lb    = (const float*)d_in[13];

    const int N    = in_sizes[0] / 64;
    const int E    = in_sizes[1] / 2;
    const int Etot = E + N;                            // + self-loops
    const long long* esrc = ei;
    const long long* edst = ei + E;

    // workspace (fp32): ~N*276*4 + 98 KB transposed weights (~110.6 MB total)
    float* bufA = (float*)d_ws;                        // xp1 -> xp2
    float* bufB = bufA + (size_t)N * 128;              // agg1/h1 -> agg2/h2
    float* as1  = bufB + (size_t)N * 128;  float* ad1 = as1 + (size_t)N * 4;
    float* mx1  = ad1 + (size_t)N * 4;     float* dn1 = mx1 + (size_t)N * 4;
    float* as2  = dn1 + (size_t)N * 4;     float* ad2 = as2 + N;
    float* mx2  = ad2 + N;                 float* dn2 = mx2 + N;
    float* wt1  = dn2 + N;                             // [128][64]
    float* wt2  = wt1 + 64 * 128;                      // [128][128]

    // ---- layer 1 ----
    fillf_kernel<<<gblk(N * 128, 256), 256, 0, stream>>>(bufB, N * 128, 0.0f);
    fillf_kernel<<<gblk(N * 4, 256), 256, 0, stream>>>(mx1, N * 4, -3.402823466e38f);
    fillf_kernel<<<gblk(N * 4, 256), 256, 0, stream>>>(dn1, N * 4, 0.0f);
    transpose_kernel<<<gblk(64 * 128, 256), 256, 0, stream>>>(W1, wt1, 64, 128);
    transpose_kernel<<<gblk(128 * 128, 256), 256, 0, stream>>>(W2, wt2, 128, 128);

    gemm_wmma_kernel<64><<<dim3(2, N / 16), 32, 0, stream>>>(x, wt1, bufA);
    attn_dots_kernel<<<(unsigned)N, 32, 0, stream>>>(bufA, atts1, attd1, as1, ad1, 4, 32);

    edge_alpha_max_kernel<4><<<gblk((long long)Etot * 4, 256), 256, 0, stream>>>(
        esrc, edst, as1, ad1, mx1, E, Etot);
    edge_expsum_kernel<4><<<gblk((long long)Etot * 4, 256), 256, 0, stream>>>(
        esrc, edst, as1, ad1, mx1, dn1, E, Etot);
    edge_agg_kernel<4><<<gblk((long long)Etot * 32, 256), 256, 0, stream>>>(
        esrc, edst, as1, ad1, mx1, dn1, bufA, bufB, E, Etot);

    bias_elu_kernel<<<gblk(N * 128, 256), 256, 0, stream>>>(bufB, b1, N * 128);

    // ---- layer 2 ----
    gemm_wmma_kernel<128><<<dim3(2, N / 16), 32, 0, stream>>>(bufB, wt2, bufA);
    attn_dots_kernel<<<(unsigned)N, 32, 0, stream>>>(bufA, atts2, attd2, as2, ad2, 1, 128);

    fillf_kernel<<<gblk(N, 256), 256, 0, stream>>>(mx2, N, -3.402823466e38f);
    fillf_kernel<<<gblk(N, 256), 256, 0, stream>>>(dn2, N, 0.0f);
    fillf_kernel<<<gblk(N * 128, 256), 256, 0, stream>>>(bufB, N * 128, 0.0f);

    edge_alpha_max_kernel<1><<<gblk(Etot, 256), 256, 0, stream>>>(
        esrc, edst, as2, ad2, mx2, E, Etot);
    edge_expsum_kernel<1><<<gblk(Etot, 256), 256, 0, stream>>>(
        esrc, edst, as2, ad2, mx2, dn2, E, Etot);
    edge_agg_kernel<1><<<gblk((long long)Etot * 32, 256), 256, 0, stream>>>(
        esrc, edst, as2, ad2, mx2, dn2, bufA, bufB, E, Etot);

    // ---- TCN (WMMA) + LayerNorm, b2 folded into the im2col load ----
    tcn_ln_kernel<<<(unsigned)(N / 16), 128, 0, stream>>>(
        bufB, b2, tw, tb, lg, lb, (float*)d_out);
}